// SplineLayer_37907381355032
// MI455X (gfx1250) — compile-verified
//
#include <hip/hip_runtime.h>
#include <math.h>

// ---------------------------------------------------------------------------
// Rational (linear-lambda) spline forward, B=65536 x D=256, K=8 bins.
//
// Kernel 1 (spline_prep): per-dimension parameter preprocessing into a packed
// coefficient table in d_ws. Per (d,bin) we store 3 float4s so the hot loop is:
//   theta = (x - icw) * inv_iw
//   left  : num = fma(theta, nL1, nL0), den = fma(theta, dL1, il)
//   right : num = fma(theta, nR1, nR0), den = fma(theta, dR1, dR0)
//   out   = num/den  (rcp + 1 Newton step)
//
// Kernel 2 (spline_main): stages the table (one D-half, 53KB) into LDS with
// CDNA5 async global->LDS b128 copies, then streams x with Δd=1 lane mapping
// (coalesced 128B VMEM, conflict-light LDS gathers).
// ---------------------------------------------------------------------------

#define DTOT   256
#define KBINS  8
#define DHALF  128
#define Q0F4   224            // 7*128 knot floats = 224 float4
#define QARRF4 1024           // 8 bins * 128 dloc float4 per coeff array
#define TBLF4  3328           // 224 + 3*1024 = 3296, padded to 13*256
#define USE_ASYNC_STAGE 1

__global__ void spline_prep(const float* __restrict__ uw,
                            const float* __restrict__ uh,
                            const float* __restrict__ ud,
                            const float* __restrict__ ul,
                            float* __restrict__ ws) {
    const int dd = threadIdx.x;            // one thread per dimension d
    if (dd >= DTOT) return;

    float w[KBINS], h[KBINS], dv[KBINS + 1], lm[KBINS];

    // ---- softmax(widths), softmax(heights) with min-bin floors ----
    float mw = -1e30f, mh = -1e30f;
    for (int k = 0; k < KBINS; ++k) {
        w[k] = uw[dd * KBINS + k];
        h[k] = uh[dd * KBINS + k];
        mw = fmaxf(mw, w[k]);
        mh = fmaxf(mh, h[k]);
    }
    float sw = 0.f, sh = 0.f;
    for (int k = 0; k < KBINS; ++k) {
        w[k] = __expf(w[k] - mw); sw += w[k];
        h[k] = __expf(h[k] - mh); sh += h[k];
    }
    const float iswn = 1.f / sw, ishn = 1.f / sh;
    const float scale = 1.0f - 1e-3f * KBINS;
    for (int k = 0; k < KBINS; ++k) {
        w[k] = 1e-3f + scale * (w[k] * iswn);
        h[k] = 1e-3f + scale * (h[k] * ishn);
    }

    // ---- knots (cumsum scaled to [-3,3], endpoints clamped exactly) ----
    float xk[KBINS + 1], yk[KBINS + 1];
    xk[0] = -3.0f; yk[0] = -3.0f;
    float cw = 0.f, ch = 0.f;
    for (int k = 0; k < KBINS; ++k) {
        cw += w[k]; xk[k + 1] = 6.0f * cw - 3.0f;
        ch += h[k]; yk[k + 1] = 6.0f * ch - 3.0f;
    }
    xk[KBINS] = 3.0f; yk[KBINS] = 3.0f;

    // ---- derivatives: softplus + pad ends with (1 - MIN_DERIV) ----
    dv[0] = 1.0f - 1e-3f; dv[KBINS] = 1.0f - 1e-3f;
    for (int k = 0; k < KBINS - 1; ++k) {
        float v = ud[dd * (KBINS - 1) + k];
        float sp = (v > 20.f) ? v : log1pf(__expf(v));
        dv[k + 1] = 1e-3f + sp;
    }
    // ---- lambdas: squashed sigmoid ----
    for (int k = 0; k < KBINS; ++k) {
        float v = ul[dd * KBINS + k];
        lm[k] = 0.95f / (1.0f + __expf(-v)) + 0.025f;
    }

    const int half = dd >> 7, dloc = dd & (DHALF - 1);
    float* wsH = ws + (size_t)half * (TBLF4 * 4);
    // interior knots xk[1..7] -> [7][128]
    for (int k = 0; k < 7; ++k) wsH[k * DHALF + dloc] = xk[k + 1];

    float4* q = ((float4*)ws) + (size_t)half * TBLF4 + Q0F4;
    for (int bin = 0; bin < KBINS; ++bin) {
        const float iw = xk[bin + 1] - xk[bin];
        const float ih = yk[bin + 1] - yk[bin];
        const float idelta = ih / iw;
        const float id0 = dv[bin], id1 = dv[bin + 1];
        const float il = lm[bin];
        const float wb = sqrtf(id0 / id1);
        const float wc = (il * id0 + (1.0f - il) * wb * id1) / idelta;
        const float ya = yk[bin], yb = yk[bin + 1];
        const float yc = ((1.0f - il) * ya + il * wb * yb) /
                         ((1.0f - il) + il * wb);
        const float nL0 = il * ya;
        const float nL1 = wc * yc - ya;
        const float dL1 = wc - 1.0f;
        const float nR0 = wc * yc - il * wb * yb;
        const float nR1 = wb * yb - wc * yc;
        const float dR0 = wc - il * wb;
        const float dR1 = wb - wc;
        q[0 * QARRF4 + bin * DHALF + dloc] = make_float4(xk[bin], 1.0f / iw, il, nL0);
        q[1 * QARRF4 + bin * DHALF + dloc] = make_float4(nL1, dL1, nR0, nR1);
        q[2 * QARRF4 + bin * DHALF + dloc] = make_float4(dR0, dR1, 0.f, 0.f);
    }
    // zero the staged padding region so LDS never sees poison (cosmetic)
    if (dd < 64) {
        const int hh = dd >> 5, ii = dd & 31;
        ((float4*)ws)[(size_t)hh * TBLF4 + 3296 + ii] = make_float4(0.f, 0.f, 0.f, 0.f);
    }
}

__global__ __launch_bounds__(256) void spline_main(const float* __restrict__ x,
                                                   float* __restrict__ out,
                                                   const float* __restrict__ ws,
                                                   int B) {
    __shared__ __align__(16) float tbl[TBLF4 * 4];   // 53,248 bytes
    const int tid  = threadIdx.x;
    const int half = blockIdx.x & 1;

    // ---- stage this D-half's table into LDS ----
    const float4* g4 = ((const float4*)ws) + (size_t)half * TBLF4;
#if USE_ASYNC_STAGE
    {
        const unsigned lbase = (unsigned)(size_t)(&tbl[0]); // LDS byte offset
        #pragma unroll
        for (int j = 0; j < 13; ++j) {
            const int f4i = tid + j * 256;
            const unsigned la = lbase + (unsigned)(f4i * 16);
            const float4* gp = g4 + f4i;
            asm volatile("global_load_async_to_lds_b128 %0, %1, off"
                         :: "v"(la), "v"(gp) : "memory");
        }
        asm volatile("s_wait_asynccnt 0" ::: "memory");
    }
#else
    {
        float4* l4 = (float4*)tbl;
        #pragma unroll
        for (int j = 0; j < 13; ++j) {
            const int f4i = tid + j * 256;
            l4[f4i] = g4[f4i];
        }
    }
#endif
    __syncthreads();

    const int dloc = tid & (DHALF - 1);
    const int rg   = tid >> 7;                 // row group 0/1
    const int d    = half * DHALF + dloc;

    // interior knots for this column, kept in registers for all rows
    float kn[7];
    #pragma unroll
    for (int k = 0; k < 7; ++k) kn[k] = tbl[k * DHALF + dloc];

    const float4* t4 = (const float4*)tbl;
    const int ntiles = B >> 3;                 // 8 rows per tile
    for (int t = (int)(blockIdx.x >> 1); t < ntiles; t += (int)(gridDim.x >> 1)) {
        const int b0 = t * 8 + rg * 4;
        #pragma unroll
        for (int j = 0; j < 4; ++j) {
            const int b = b0 + j;
            const float xv = x[(size_t)b * DTOT + d];

            const bool inside = (xv >= -3.0f) && (xv <= 3.0f);
            const float xin = inside ? xv : 0.0f;

            int bin = 0;
            #pragma unroll
            for (int k = 0; k < 7; ++k) bin += (xin >= kn[k]) ? 1 : 0;

            const int qi = Q0F4 + bin * DHALF + dloc;
            const float4 q0 = t4[qi];
            const float4 q1 = t4[qi + QARRF4];
            const float4 q2 = t4[qi + 2 * QARRF4];

            const float theta = (xin - q0.x) * q0.y;
            const bool left = theta <= q0.z;
            const float num = left ? fmaf(theta, q1.x, q0.w)
                                   : fmaf(theta, q1.w, q1.z);
            const float den = left ? fmaf(theta, q1.y, q0.z)
                                   : fmaf(theta, q2.y, q2.x);
            float r = __builtin_amdgcn_rcpf(den);
            r = r * fmaf(-den, r, 2.0f);       // one Newton step
            const float o = num * r;

            out[(size_t)b * DTOT + d] = inside ? o : xv;
        }
    }
}

extern "C" void kernel_launch(void* const* d_in, const int* in_sizes, int n_in,
                              void* d_out, int out_size, void* d_ws, size_t ws_size,
                              hipStream_t stream) {
    const float* x  = (const float*)d_in[0];
    const float* uw = (const float*)d_in[1];
    const float* uh = (const float*)d_in[2];
    const float* ud = (const float*)d_in[3];
    const float* ul = (const float*)d_in[4];
    float* out = (float*)d_out;
    float* ws  = (float*)d_ws;

    const int n = in_sizes[0];     // B * D
    const int B = n / DTOT;        // 65536

    spline_prep<<<1, 256, 0, stream>>>(uw, uh, ud, ul, ws);

    // 2048 blocks (even count: blocks pair up over the two D-halves),
    // grid-stride over 8-row tiles.
    spline_main<<<2048, 256, 0, stream>>>(x, out, ws, B);
}